// CenterSeperateMarginLoss_15298673508618
// MI455X (gfx1250) — compile-verified
//
#include <hip/hip_runtime.h>
#include <math.h>

typedef float v2f __attribute__((ext_vector_type(2)));
typedef float v8f __attribute__((ext_vector_type(8)));

#define B_ROWS    32768
#define N_CLASSES 1000
#define C_PAD     1024
#define D_FEATS   128
#define MARGIN_F   0.25f
#define DISTANCE_F 1.0f
#define EMA_DECAY_F 0.999f

// ---- workspace layout (float offsets) ----
#define WS_SUMS    0                         // 1000*128 = 128000
#define WS_COUNTS  128000                    // 1000
#define WS_ACCUM   129000                    // 4 (sum_p, cnt_p, sum_n, cnt_n)
#define WS_XN      129004                    // 32768
#define WS_CENTERS 161772                    // 1024*128 = 131072
#define WS_CN      292844                    // 1024
#define WS_ZERO_N  129004                    // region to clear each launch

__global__ void zero_kernel(float* __restrict__ p, int n) {
  int i = blockIdx.x * 256 + threadIdx.x;
  if (i < n) p[i] = 0.0f;
}

// one thread per (row, feat): atomic segment-sum into sums[label][feat]
__global__ void segsum_kernel(const float* __restrict__ x,
                              const int* __restrict__ labels,
                              float* __restrict__ sums,
                              float* __restrict__ counts) {
  int idx = blockIdx.x * 256 + threadIdx.x;        // B_ROWS*D_FEATS threads
  int b = idx >> 7;
  int d = idx & 127;
  int lab = labels[b];
  atomicAdd(&sums[lab * D_FEATS + d], x[idx]);
  if (d == 0) atomicAdd(&counts[lab], 1.0f);
}

// one block (128 thr) per row: xn[row] = sum(x[row]^2)
__global__ void xnorm_kernel(const float* __restrict__ x, float* __restrict__ xn) {
  int row = blockIdx.x;
  float v = x[row * D_FEATS + threadIdx.x];
  float sq = v * v;
  for (int off = 16; off > 0; off >>= 1) sq += __shfl_down(sq, off, 32);
  __shared__ float part[4];
  if ((threadIdx.x & 31) == 0) part[threadIdx.x >> 5] = sq;
  __syncthreads();
  if (threadIdx.x == 0) xn[row] = part[0] + part[1] + part[2] + part[3];
}

// one block (128 thr) per (padded) class: EMA center + ||center||^2
__global__ void centers_kernel(const float* __restrict__ old_c,
                               const float* __restrict__ sums,
                               const float* __restrict__ counts,
                               const int* __restrict__ ema_it,
                               float* __restrict__ centers,
                               float* __restrict__ cn) {
  int c = blockIdx.x;
  int d = threadIdx.x;
  float v = 0.0f;
  if (c < N_CLASSES) {
    float it = (float)ema_it[0];
    float alpha = fminf(1.0f - 1.0f / (it + 1.0f), EMA_DECAY_F);
    float cnt = counts[c];
    float oldv = old_c[c * D_FEATS + d];
    float bm = (cnt > 0.0f) ? (sums[c * D_FEATS + d] / fmaxf(cnt, 1.0f)) : oldv;
    v = alpha * oldv + (1.0f - alpha) * bm;
  }
  centers[c * D_FEATS + d] = v;
  float sq = v * v;
  for (int off = 16; off > 0; off >>= 1) sq += __shfl_down(sq, off, 32);
  __shared__ float part[4];
  if ((threadIdx.x & 31) == 0) part[threadIdx.x >> 5] = sq;
  __syncthreads();
  if (threadIdx.x == 0) cn[c] = part[0] + part[1] + part[2] + part[3];
}

// main: each wave computes a 16x64 tile of x @ centers^T via V_WMMA_F32_16X16X4_F32,
// then folds distances into the margin-loss partial sums.
__global__ void __launch_bounds__(256)
dist_loss_kernel(const float* __restrict__ x,
                 const int* __restrict__ labels,
                 const float* __restrict__ centers,
                 const float* __restrict__ xn,
                 const float* __restrict__ cn,
                 float* __restrict__ accum) {
  const int lane = threadIdx.x & 31;
  const int wave = threadIdx.x >> 5;
  const int gw = blockIdx.x * 8 + wave;      // 32768 waves total
  const int rowTile = gw >> 4;               // 2048 row tiles of 16
  const int colT = gw & 15;                  // 16 col groups of 64
  const int rowBase = rowTile * 16;
  const int colBase = colT * 64;
  const int m  = lane & 15;                  // row/col within 16
  const int kh = lane >> 4;                  // K half-select (0 or 1)

  // A: 16x4 f32 tile of x; B: 4x16 f32 tile of centers^T. Per ISA layout,
  // lane L reads a float2 at k = kk + 2*(L>>4) for row/col (L&15).
  const float* xr = x + (size_t)(rowBase + m) * D_FEATS + 2 * kh;
  const float* c0 = centers + (size_t)(colBase +  0 + m) * D_FEATS + 2 * kh;
  const float* c1 = centers + (size_t)(colBase + 16 + m) * D_FEATS + 2 * kh;
  const float* c2 = centers + (size_t)(colBase + 32 + m) * D_FEATS + 2 * kh;
  const float* c3 = centers + (size_t)(colBase + 48 + m) * D_FEATS + 2 * kh;

  v8f acc[4];
  acc[0] = (v8f)0.0f; acc[1] = (v8f)0.0f; acc[2] = (v8f)0.0f; acc[3] = (v8f)0.0f;

  for (int kk = 0; kk < D_FEATS; kk += 4) {
    v2f a  = *(const v2f*)(xr + kk);
    v2f b0 = *(const v2f*)(c0 + kk);
    v2f b1 = *(const v2f*)(c1 + kk);
    v2f b2 = *(const v2f*)(c2 + kk);
    v2f b3 = *(const v2f*)(c3 + kk);
    acc[0] = __builtin_amdgcn_wmma_f32_16x16x4_f32(false, a, false, b0, (short)0, acc[0], false, false);
    acc[1] = __builtin_amdgcn_wmma_f32_16x16x4_f32(false, a, false, b1, (short)0, acc[1], false, false);
    acc[2] = __builtin_amdgcn_wmma_f32_16x16x4_f32(false, a, false, b2, (short)0, acc[2], false, false);
    acc[3] = __builtin_amdgcn_wmma_f32_16x16x4_f32(false, a, false, b3, (short)0, acc[3], false, false);
  }

  // C/D layout: VGPR r, lanes 0-15 -> row rowBase+r, lanes 16-31 -> row rowBase+r+8; col = colBase+16t+(L&15)
  const int rowL = rowBase + 8 * kh;
  int   lab[8];
  float xnr[8];
#pragma unroll
  for (int r = 0; r < 8; ++r) {
    lab[r] = labels[rowL + r];
    xnr[r] = xn[rowL + r];
  }

  float sp = 0.0f, cp = 0.0f, sn = 0.0f, cnn = 0.0f;
#pragma unroll
  for (int t = 0; t < 4; ++t) {
    const int col = colBase + t * 16 + m;
    const bool valid = (col < N_CLASSES);
    const float cnv = cn[col];
#pragma unroll
    for (int r = 0; r < 8; ++r) {
      float dot = acc[t][r];
      float sq = xnr[r] + cnv - 2.0f * dot;
      float delta = sqrtf(fmaxf(sq, 1e-12f));
      if (valid) {
        if (lab[r] == col) {
          // ap*ps = (ps + DISTANCE)*ps  since ps >= 0
          float ps = fmaxf(delta - MARGIN_F, 0.0f);
          sp += (ps + DISTANCE_F) * ps;
          cp += (ps > 0.0f) ? 1.0f : 0.0f;
        } else {
          // an*ns = (ns + MARGIN)*ns  since ns >= 0
          float ns = fmaxf(DISTANCE_F - delta, 0.0f);
          sn += (ns + MARGIN_F) * ns;
          cnn += (ns > 0.0f) ? 1.0f : 0.0f;
        }
      }
    }
  }

  for (int off = 16; off > 0; off >>= 1) {
    sp  += __shfl_down(sp,  off, 32);
    cp  += __shfl_down(cp,  off, 32);
    sn  += __shfl_down(sn,  off, 32);
    cnn += __shfl_down(cnn, off, 32);
  }
  if (lane == 0) {
    atomicAdd(&accum[0], sp);
    atomicAdd(&accum[1], cp);
    atomicAdd(&accum[2], sn);
    atomicAdd(&accum[3], cnn);
  }
}

__global__ void finalize_kernel(const float* __restrict__ accum, float* __restrict__ out) {
  if (blockIdx.x == 0 && threadIdx.x == 0) {
    float lp = accum[0] / (accum[1] + 1.0f);
    float ln = accum[2] / (accum[3] + 1.0f);
    out[0] = logf(1.0f + ln + lp);
  }
}

extern "C" void kernel_launch(void* const* d_in, const int* in_sizes, int n_in,
                              void* d_out, int out_size, void* d_ws, size_t ws_size,
                              hipStream_t stream) {
  const float* x      = (const float*)d_in[0];
  const float* old_c  = (const float*)d_in[1];
  const int*   labels = (const int*)d_in[2];
  const int*   ema_it = (const int*)d_in[3];
  float* out = (float*)d_out;

  float* ws      = (float*)d_ws;
  float* sums    = ws + WS_SUMS;
  float* counts  = ws + WS_COUNTS;
  float* accum   = ws + WS_ACCUM;
  float* xn      = ws + WS_XN;
  float* centers = ws + WS_CENTERS;
  float* cn      = ws + WS_CN;

  // 1) clear segment sums / counts / loss accumulators
  zero_kernel<<<(WS_ZERO_N + 255) / 256, 256, 0, stream>>>(ws, WS_ZERO_N);
  // 2) segment sum by label
  segsum_kernel<<<(B_ROWS * D_FEATS) / 256, 256, 0, stream>>>(x, labels, sums, counts);
  // 3) per-row squared norms
  xnorm_kernel<<<B_ROWS, D_FEATS, 0, stream>>>(x, xn);
  // 4) EMA centers (padded to 1024) + per-class squared norms
  centers_kernel<<<C_PAD, D_FEATS, 0, stream>>>(old_c, sums, counts, ema_it, centers, cn);
  // 5) WMMA distance + margin-loss partial reduction
  dist_loss_kernel<<<(B_ROWS / 16) * (C_PAD / 64) / 8, 256, 0, stream>>>(
      x, labels, centers, xn, cn, accum);
  // 6) scalar finalize
  finalize_kernel<<<1, 1, 0, stream>>>(accum, out);
}